// Graphsage_27470610825589
// MI455X (gfx1250) — compile-verified
//
#include <hip/hip_runtime.h>
#include <hip/hip_bf16.h>

typedef float v2f __attribute__((ext_vector_type(2)));
typedef float v8f __attribute__((ext_vector_type(8)));

#define FDIM 128

// ---------------- utility kernels ----------------
__global__ __launch_bounds__(256) void k_zero(float* __restrict__ p, long long n) {
  long long i = (long long)blockIdx.x * blockDim.x + threadIdx.x;
  if (i < n) p[i] = 0.0f;
}

__global__ __launch_bounds__(256) void k_init_logits(float* __restrict__ out,
                                                     const float* __restrict__ blin, int n) {
  int i = blockIdx.x * blockDim.x + threadIdx.x;
  if (i < 2 * n) out[i] = blin[i & 1];
}

__global__ __launch_bounds__(256) void k_degree(const long long* __restrict__ dst,
                                                float* __restrict__ deg, int nE) {
  int e = blockIdx.x * blockDim.x + threadIdx.x;
  if (e < nE) atomicAdd(&deg[(int)dst[e]], 1.0f);
}

// 32 threads per edge; each thread moves 4 consecutive floats (float4 gather,
// 4x global_atomic_add_f32 scatter). Whole agg buffer is L2-resident.
__global__ __launch_bounds__(256) void k_scatter(const float* __restrict__ feat,
                                                 const long long* __restrict__ src,
                                                 const long long* __restrict__ dst,
                                                 float* __restrict__ agg, int nE) {
  long long tid = (long long)blockIdx.x * blockDim.x + threadIdx.x;
  int e = (int)(tid >> 5);
  if (e >= nE) return;
  int f0 = ((int)tid & 31) << 2;
  int s = (int)src[e], d = (int)dst[e];
  const float4 v = *(const float4*)(feat + (size_t)s * FDIM + f0);
  float* a = agg + (size_t)d * FDIM + f0;
  atomicAdd(a + 0, v.x);
  atomicAdd(a + 1, v.y);
  atomicAdd(a + 2, v.z);
  atomicAdd(a + 3, v.w);
}

// ---------------- fused SAGE layer: h = relu((agg/deg) @ Wl^T + bl + xroot @ Wr^T) ----------------
// One wave computes a 16(nodes) x 16(out-feats) tile with V_WMMA_F32_16X16X4_F32.
// Optional fused head: accumulate relu(h) @ Wlin^T into logits via half-wave shfl reduction.
__global__ __launch_bounds__(256) void k_sage_gemm(
    const float* __restrict__ agg, const float* __restrict__ deg,
    const float* __restrict__ xroot,
    const float* __restrict__ Wl, const float* __restrict__ bl,
    const float* __restrict__ Wr,
    float* __restrict__ hout,        // nullable
    float* __restrict__ logits,      // nullable
    const float* __restrict__ Wlin,  // [2, FDIM], used iff logits != nullptr
    int nNodes) {
  const int lane = threadIdx.x & 31;
  const int wave = threadIdx.x >> 5;
  const int m0 = blockIdx.x * 16;      // node tile base (nNodes % 16 == 0)
  const int n0 = wave * 16;            // out-feature tile base (8 waves cover 128)
  const int r  = lane & 15;            // row (A) / col (B,D) within tile
  const int ko = (lane >> 4) << 1;     // K sub-offset: lanes 0-15 -> K+0/K+1, lanes 16-31 -> K+2/K+3
  if (m0 >= nNodes) return;

  const int row = m0 + r;
  const float invd = 1.0f / fmaxf(deg[row], 1.0f);
  const float* aggRow = agg   + (size_t)row * FDIM;
  const float* xRow   = xroot + (size_t)row * FDIM;
  const float* wlRow  = Wl + (size_t)(n0 + r) * FDIM;
  const float* wrRow  = Wr + (size_t)(n0 + r) * FDIM;

  v8f acc = {0.f, 0.f, 0.f, 0.f, 0.f, 0.f, 0.f, 0.f};
  for (int k = 0; k < FDIM; k += 4) {
    v2f a, b;
    // neighbor-aggregate branch: (agg/deg) @ Wl^T
    a.x = aggRow[k + ko] * invd;  a.y = aggRow[k + ko + 1] * invd;
    b.x = wlRow[k + ko];          b.y = wlRow[k + ko + 1];
    acc = __builtin_amdgcn_wmma_f32_16x16x4_f32(false, a, false, b, (short)0, acc,
                                                false, false);
    // root branch: x @ Wr^T
    a.x = xRow[k + ko];           a.y = xRow[k + ko + 1];
    b.x = wrRow[k + ko];          b.y = wrRow[k + ko + 1];
    acc = __builtin_amdgcn_wmma_f32_16x16x4_f32(false, a, false, b, (short)0, acc,
                                                false, false);
  }

  const int col = n0 + r;
  const float bias = bl[col];
  const int mhalf = (lane >> 4) << 3;  // D layout: lanes 16-31 hold rows M+8

  if (hout) {
#pragma unroll
    for (int v = 0; v < 8; ++v) {
      int m = m0 + v + mhalf;
      hout[(size_t)m * FDIM + col] = fmaxf(acc[v] + bias, 0.0f);
    }
  }
  if (logits) {
    const float w0 = Wlin[col];
    const float w1 = Wlin[FDIM + col];
#pragma unroll
    for (int v = 0; v < 8; ++v) {
      int m = m0 + v + mhalf;
      float h = fmaxf(acc[v] + bias, 0.0f);
      float p0 = h * w0, p1 = h * w1;
      // reduce across the 16 lanes of this half-wave (all share the same m)
#pragma unroll
      for (int off = 8; off >= 1; off >>= 1) {
        p0 += __shfl_xor(p0, off, 16);
        p1 += __shfl_xor(p1, off, 16);
      }
      if (r == 0) {
        atomicAdd(&logits[(size_t)m * 2 + 0], p0);
        atomicAdd(&logits[(size_t)m * 2 + 1], p1);
      }
    }
  }
}

__global__ __launch_bounds__(256) void k_logsoftmax(float* __restrict__ out, int n) {
  int i = blockIdx.x * blockDim.x + threadIdx.x;
  if (i >= n) return;
  float a = out[2 * i], b = out[2 * i + 1];
  float m = fmaxf(a, b);
  float lse = m + logf(expf(a - m) + expf(b - m));
  out[2 * i]     = a - lse;
  out[2 * i + 1] = b - lse;
}

// ---------------- launcher ----------------
extern "C" void kernel_launch(void* const* d_in, const int* in_sizes, int n_in,
                              void* d_out, int out_size, void* d_ws, size_t ws_size,
                              hipStream_t stream) {
  (void)n_in; (void)out_size; (void)ws_size;
  const float*     x    = (const float*)d_in[0];
  const long long* ei   = (const long long*)d_in[1];
  const float*     W1l  = (const float*)d_in[2];
  const float*     b1l  = (const float*)d_in[3];
  const float*     W1r  = (const float*)d_in[4];
  const float*     W2l  = (const float*)d_in[5];
  const float*     b2l  = (const float*)d_in[6];
  const float*     W2r  = (const float*)d_in[7];
  const float*     Wlin = (const float*)d_in[8];
  const float*     blin = (const float*)d_in[9];

  const int N = in_sizes[0] / FDIM;   // 50000 (divisible by 16)
  const int E = in_sizes[1] / 2;      // 800000
  const long long* srcIdx = ei;       // edge_index[0]
  const long long* dstIdx = ei + E;   // edge_index[1]

  float* deg = (float*)d_ws;                     // N floats
  float* agg = deg + N;                          // N*FDIM floats
  float* h1  = agg + (size_t)N * FDIM;           // N*FDIM floats
  float* logits = (float*)d_out;                 // N*2 floats

  dim3 blk(256);
  const long long zAll = (long long)N + (long long)N * FDIM;  // deg+agg contiguous
  const long long zAgg = (long long)N * FDIM;
  const long long sThreads = (long long)E * 32;

  // ---- layer 1 ----
  k_zero<<<dim3((unsigned)((zAll + 255) / 256)), blk, 0, stream>>>(deg, zAll);
  k_init_logits<<<dim3((2 * N + 255) / 256), blk, 0, stream>>>(logits, blin, N);
  k_degree<<<dim3((E + 255) / 256), blk, 0, stream>>>(dstIdx, deg, E);
  k_scatter<<<dim3((unsigned)((sThreads + 255) / 256)), blk, 0, stream>>>(x, srcIdx, dstIdx, agg, E);
  k_sage_gemm<<<dim3(N / 16), blk, 0, stream>>>(agg, deg, x, W1l, b1l, W1r,
                                                h1, nullptr, nullptr, N);
  // ---- layer 2 (fused classifier head) ----
  k_zero<<<dim3((unsigned)((zAgg + 255) / 256)), blk, 0, stream>>>(agg, zAgg);
  k_scatter<<<dim3((unsigned)((sThreads + 255) / 256)), blk, 0, stream>>>(h1, srcIdx, dstIdx, agg, E);
  k_sage_gemm<<<dim3(N / 16), blk, 0, stream>>>(agg, deg, h1, W2l, b2l, W2r,
                                                nullptr, logits, Wlin, N);
  k_logsoftmax<<<dim3((N + 255) / 256), blk, 0, stream>>>(logits, N);
}